// Net_1967095022180
// MI455X (gfx1250) — compile-verified
//
#include <hip/hip_runtime.h>
#include <hip/hip_bf16.h>

// ---------------------------------------------------------------------------
// Problem constants (match reference)
// ---------------------------------------------------------------------------
constexpr int NN  = 10000;           // nodes
constexpr int BB  = 32;              // batch
constexpr int EE  = 40000;           // edges
constexpr int HH  = 4;               // heads
constexpr int DD  = 10;              // feats per head
constexpr int TT  = 10;              // input feats
constexpr int NB  = NN * BB;         // 320000 rows
constexpr int LNM = NB * DD;         // 3,200,000 elems per (N,B,D) tensor
constexpr int FTC = HH * DD + 2*HH;  // 48 = 40 ft cols + 4 el + 4 er
constexpr float NEG = 0.2f;

typedef float v2f __attribute__((ext_vector_type(2)));
typedef float v8f __attribute__((ext_vector_type(8)));

__device__ __forceinline__ float leaky(float v) { return v >= 0.0f ? v : NEG * v; }

// monotonic float<->uint key for atomicMax on signed floats
__device__ __forceinline__ unsigned fkey(float f) {
    unsigned u = __float_as_uint(f);
    return (u & 0x80000000u) ? ~u : (u | 0x80000000u);
}
__device__ __forceinline__ float fdec(unsigned k) {
    unsigned u = (k & 0x80000000u) ? (k & 0x7fffffffu) : ~k;
    return __uint_as_float(u);
}
__device__ __forceinline__ void atomAddF(float* p, float v) {
    __hip_atomic_fetch_add(p, v, __ATOMIC_RELAXED, __HIP_MEMORY_SCOPE_AGENT);
}

// ---------------------------------------------------------------------------
// 0) vectorized zero (n4 = count of float4)
// ---------------------------------------------------------------------------
__global__ __launch_bounds__(256) void k_zero4(float4* __restrict__ p, int n4) {
    int tid = blockIdx.x * blockDim.x + threadIdx.x;
    if (tid < n4) p[tid] = make_float4(0.f, 0.f, 0.f, 0.f);
}

// ---------------------------------------------------------------------------
// 1) unpack x (T,N,B,3) -> mu (N,B,T), sigma (N,B,T)       [3.2M threads]
// ---------------------------------------------------------------------------
__global__ __launch_bounds__(256) void k_unpack(const float* __restrict__ x,
                                                float* __restrict__ mu,
                                                float* __restrict__ sg) {
    int tid = blockIdx.x * blockDim.x + threadIdx.x;   // exact: 12500*256 == NB*TT
    int r = tid / TT, t = tid - r * TT;
    int n = r >> 5, b = r & 31;
    int xi = ((t * NN + n) * BB + b) * 3;
    mu[tid] = x[xi];
    sg[tid] = x[xi + 1];
}

// ---------------------------------------------------------------------------
// 2) build combined B matrix (12 x 48): cols 0..39 = W^T, 40..43 = W^T a_l,
//    44..47 = W^T a_r. Rows 10,11 are zero padding. Also zero red[0..1].
// ---------------------------------------------------------------------------
__global__ void k_prep_w(const float* __restrict__ fc,   // (40,10)
                         const float* __restrict__ al,   // (4,10)
                         const float* __restrict__ ar,   // (4,10)
                         float* __restrict__ Bmat,       // (12,48)
                         float* __restrict__ red) {
    if (threadIdx.x < 2) red[threadIdx.x] = 0.0f;
    for (int i = threadIdx.x; i < 12 * FTC; i += blockDim.x) {
        int t = i / FTC, o = i - t * FTC;
        float v = 0.0f;
        if (t < TT) {
            if (o < HH * DD) {
                v = fc[o * TT + t];
            } else if (o < HH * DD + HH) {
                int h = o - HH * DD;
                float s = 0.0f;
                for (int d = 0; d < DD; ++d) s += fc[(h * DD + d) * TT + t] * al[h * DD + d];
                v = s;
            } else {
                int h = o - HH * DD - HH;
                float s = 0.0f;
                for (int d = 0; d < DD; ++d) s += fc[(h * DD + d) * TT + t] * ar[h * DD + d];
                v = s;
            }
        }
        Bmat[i] = v;
    }
}

// ---------------------------------------------------------------------------
// 3) WMMA GEMM: (NB x 10) * (10 x 48) -> ft (NB x 48)
//    One 16-row tile per wave32; 3 col-tiles x 3 k-steps of V_WMMA_F32_16X16X4_F32.
//    A tile loaded coalesced (160 contiguous floats); D tile staged through LDS
//    and written as 6 coalesced b128 stores per lane (tile is a contiguous
//    768-float span of ft). Grid 5000 x 128 (4 waves) = 20000 row tiles exact.
// ---------------------------------------------------------------------------
__global__ __launch_bounds__(128) void k_gemm(const float* __restrict__ feat,
                                              const float* __restrict__ Bmat,
                                              float* __restrict__ ft) {
    __shared__ float Abuf[4][16][12];
    __shared__ __align__(16) float Obuf[4][16 * FTC];   // 768 floats per wave
    const int wave = threadIdx.x >> 5;
    const int lane = threadIdx.x & 31;
    const int r0   = (blockIdx.x * 4 + wave) * 16;      // row-tile base, exact

    // stage A tile: 160 contiguous floats, coalesced; pad cols 10..11 with 0
    const float* fbase = feat + (size_t)r0 * TT;
    #pragma unroll
    for (int i = 0; i < 5; ++i) {
        int t = lane + i * 32;                 // 0..159
        int row = t / TT, col = t - row * TT;
        Abuf[wave][row][col] = fbase[t];
    }
    Abuf[wave][lane >> 1][10 + (lane & 1)] = 0.0f;
    __syncthreads();

    const int m   = lane & 15;                // M row (A) / N col (B,D)
    const int khi = (lane < 16) ? 0 : 2;      // K sub-offset per ISA fragment layout

    v8f acc0 = {0,0,0,0,0,0,0,0};
    v8f acc1 = {0,0,0,0,0,0,0,0};
    v8f acc2 = {0,0,0,0,0,0,0,0};

    #pragma unroll
    for (int k0 = 0; k0 < 12; k0 += 4) {
        v2f a;
        a.x = Abuf[wave][m][k0 + khi + 0];
        a.y = Abuf[wave][m][k0 + khi + 1];
        v2f b0, b1, b2;
        b0.x = Bmat[(k0 + khi + 0) * FTC +  0 + m];
        b0.y = Bmat[(k0 + khi + 1) * FTC +  0 + m];
        b1.x = Bmat[(k0 + khi + 0) * FTC + 16 + m];
        b1.y = Bmat[(k0 + khi + 1) * FTC + 16 + m];
        b2.x = Bmat[(k0 + khi + 0) * FTC + 32 + m];
        b2.y = Bmat[(k0 + khi + 1) * FTC + 32 + m];
        acc0 = __builtin_amdgcn_wmma_f32_16x16x4_f32(false, a, false, b0, (short)0, acc0, false, false);
        acc1 = __builtin_amdgcn_wmma_f32_16x16x4_f32(false, a, false, b1, (short)0, acc1, false, false);
        acc2 = __builtin_amdgcn_wmma_f32_16x16x4_f32(false, a, false, b2, (short)0, acc2, false, false);
    }

    // stage D tile into LDS, then write out coalesced b128
    const int rb = (lane < 16) ? 0 : 8;
    #pragma unroll
    for (int v = 0; v < 8; ++v) {
        Obuf[wave][(rb + v) * FTC +  0 + m] = acc0[v];
        Obuf[wave][(rb + v) * FTC + 16 + m] = acc1[v];
        Obuf[wave][(rb + v) * FTC + 32 + m] = acc2[v];
    }
    __syncthreads();
    float4* obase = (float4*)(ft + (size_t)r0 * FTC);   // 16B aligned (r0*192 bytes)
    const float4* lsrc = (const float4*)&Obuf[wave][0];
    #pragma unroll
    for (int i = 0; i < 6; ++i) {
        obase[lane + i * 32] = lsrc[lane + i * 32];
    }
}

// ---------------------------------------------------------------------------
// 4) edge pass 1: segment-max of leaky(el[src]+er[dst]) over dst
//    one thread per (edge,batch) pair; exact grid 5000*256 = EE*BB
//    el/er rows are 16B aligned within ft (row stride 192B, offsets 160/176B)
// ---------------------------------------------------------------------------
__global__ __launch_bounds__(256) void k_edge_max(const float* __restrict__ ft,
                                                  const int* __restrict__ src,
                                                  const int* __restrict__ dst,
                                                  unsigned* __restrict__ emaxk) {
    int tid = blockIdx.x * blockDim.x + threadIdx.x;
    int e = tid >> 5, b = tid & 31;
    int sr = src[e] * BB + b, dr = dst[e] * BB + b;
    float4 el4 = *(const float4*)(ft + (size_t)sr * FTC + 40);
    float4 er4 = *(const float4*)(ft + (size_t)dr * FTC + 44);
    float ev[HH] = { leaky(el4.x + er4.x), leaky(el4.y + er4.y),
                     leaky(el4.z + er4.z), leaky(el4.w + er4.w) };
    #pragma unroll
    for (int h = 0; h < HH; ++h) atomicMax(&emaxk[dr * HH + h], fkey(ev[h]));
}

// 5) edge pass 2: esum[dst] += exp(e - emax[dst])
__global__ __launch_bounds__(256) void k_edge_sum(const float* __restrict__ ft,
                                                  const int* __restrict__ src,
                                                  const int* __restrict__ dst,
                                                  const unsigned* __restrict__ emaxk,
                                                  float* __restrict__ esum) {
    int tid = blockIdx.x * blockDim.x + threadIdx.x;
    int e = tid >> 5, b = tid & 31;
    int sr = src[e] * BB + b, dr = dst[e] * BB + b;
    float4 el4 = *(const float4*)(ft + (size_t)sr * FTC + 40);
    float4 er4 = *(const float4*)(ft + (size_t)dr * FTC + 44);
    uint4  mk  = *(const uint4*)(emaxk + (size_t)dr * HH);
    float ev[HH] = { leaky(el4.x + er4.x), leaky(el4.y + er4.y),
                     leaky(el4.z + er4.z), leaky(el4.w + er4.w) };
    unsigned mks[HH] = { mk.x, mk.y, mk.z, mk.w };
    #pragma unroll
    for (int h = 0; h < HH; ++h) {
        atomAddF(&esum[dr * HH + h], __expf(ev[h] - fdec(mks[h])));
    }
}

// 6) edge pass 3: outacc[dst,h,d] += ft[src,h,d] * a(edge,h)
__global__ __launch_bounds__(256) void k_edge_scatter(const float* __restrict__ ft,
                                                      const int* __restrict__ src,
                                                      const int* __restrict__ dst,
                                                      const unsigned* __restrict__ emaxk,
                                                      const float* __restrict__ esum,
                                                      float* __restrict__ outacc) {
    int tid = blockIdx.x * blockDim.x + threadIdx.x;
    int e = tid >> 5, b = tid & 31;
    int sr = src[e] * BB + b, dr = dst[e] * BB + b;
    float4 el4 = *(const float4*)(ft + (size_t)sr * FTC + 40);
    float4 er4 = *(const float4*)(ft + (size_t)dr * FTC + 44);
    uint4  mk  = *(const uint4*)(emaxk + (size_t)dr * HH);
    float4 es4 = *(const float4*)(esum + (size_t)dr * HH);
    float ev[HH] = { leaky(el4.x + er4.x), leaky(el4.y + er4.y),
                     leaky(el4.z + er4.z), leaky(el4.w + er4.w) };
    unsigned mks[HH] = { mk.x, mk.y, mk.z, mk.w };
    float ess[HH] = { es4.x, es4.y, es4.z, es4.w };

    // load the whole 40-float ft[src] row as 10 aligned float4
    float4 fr[10];
    const float4* fs = (const float4*)(ft + (size_t)sr * FTC);
    #pragma unroll
    for (int i = 0; i < 10; ++i) fr[i] = fs[i];
    const float* fv = (const float*)fr;

    #pragma unroll
    for (int h = 0; h < HH; ++h) {
        float a = __expf(ev[h] - fdec(mks[h])) / ess[h];
        #pragma unroll
        for (int d = 0; d < DD; ++d) {
            atomAddF(&outacc[((size_t)dr * HH + h) * DD + d], fv[h * DD + d] * a);
        }
    }
}

// ---------------------------------------------------------------------------
// 7a) head-mean of leaky(outacc) -> tmp, plus global sum/sumsq for LayerNorm
//     one thread per row; 10 aligned float4 loads. exact grid 1250*256 = NB
// ---------------------------------------------------------------------------
__global__ __launch_bounds__(256) void k_hm_reduce(const float* __restrict__ outacc,
                                                   float* __restrict__ tmp,
                                                   float* __restrict__ red) {
    int r = blockIdx.x * blockDim.x + threadIdx.x;
    float4 q[10];
    const float4* ob = (const float4*)(outacc + (size_t)r * HH * DD);  // 160B/row
    #pragma unroll
    for (int i = 0; i < 10; ++i) q[i] = ob[i];
    const float* o = (const float*)q;
    float s1 = 0.0f, s2 = 0.0f;
    #pragma unroll
    for (int d = 0; d < DD; ++d) {
        float s = 0.25f * (leaky(o[d]) + leaky(o[DD + d]) + leaky(o[2*DD + d]) + leaky(o[3*DD + d]));
        tmp[(size_t)r * DD + d] = s;
        s1 += s;
        s2 += s * s;
    }
    __shared__ float sh1[256], sh2[256];
    sh1[threadIdx.x] = s1;
    sh2[threadIdx.x] = s2;
    __syncthreads();
    for (int st = 128; st > 0; st >>= 1) {
        if (threadIdx.x < st) {
            sh1[threadIdx.x] += sh1[threadIdx.x + st];
            sh2[threadIdx.x] += sh2[threadIdx.x + st];
        }
        __syncthreads();
    }
    if (threadIdx.x == 0) {
        atomAddF(&red[0], sh1[0]);
        atomAddF(&red[1], sh2[0]);
    }
}

// 7b) whole-tensor LayerNorm apply (float4; exact grid 3125*256 = LNM/4)
__global__ __launch_bounds__(256) void k_ln_apply(const float4* __restrict__ tmp,
                                                  const float* __restrict__ red,
                                                  float4* __restrict__ dst) {
    int tid = blockIdx.x * blockDim.x + threadIdx.x;
    const float invM = 1.0f / (float)LNM;
    float m = red[0] * invM;
    float v = red[1] * invM - m * m;
    float rs = rsqrtf(v + 1e-5f);
    float4 t = tmp[tid];
    dst[tid] = make_float4((t.x - m) * rs, (t.y - m) * rs, (t.z - m) * rs, (t.w - m) * rs);
}

// 7c) final-layer GATs: add head-mean of leaky(outacc) / 3 into output half
//     one thread per row; exact grid 1250*256 = NB
__global__ __launch_bounds__(256) void k_hm_addout(const float* __restrict__ outacc,
                                                   float* __restrict__ outhalf) {
    int r = blockIdx.x * blockDim.x + threadIdx.x;
    float4 q[10];
    const float4* ob = (const float4*)(outacc + (size_t)r * HH * DD);
    #pragma unroll
    for (int i = 0; i < 10; ++i) q[i] = ob[i];
    const float* o = (const float*)q;
    #pragma unroll
    for (int d = 0; d < DD; ++d) {
        float s = 0.25f * (leaky(o[d]) + leaky(o[DD + d]) + leaky(o[2*DD + d]) + leaky(o[3*DD + d]));
        outhalf[(size_t)r * DD + d] += s * (1.0f / 3.0f);
    }
}

// ---------------------------------------------------------------------------
// 8) h = (a+b)/2   (float4; exact grid 3125*256)
// ---------------------------------------------------------------------------
__global__ __launch_bounds__(256) void k_combine(const float4* __restrict__ a,
                                                 const float4* __restrict__ b,
                                                 float4* __restrict__ h) {
    int tid = blockIdx.x * blockDim.x + threadIdx.x;
    float4 va = a[tid], vb = b[tid];
    h[tid] = make_float4(0.5f * (va.x + vb.x), 0.5f * (va.y + vb.y),
                         0.5f * (va.z + vb.z), 0.5f * (va.w + vb.w));
}

// ---------------------------------------------------------------------------
// 9) init output with broadcast regression terms /3 (overwrites poison)
//    one thread per (n,b) row; exact grid 1250*256 = NB
// ---------------------------------------------------------------------------
__global__ __launch_bounds__(256) void k_init_out(const float* __restrict__ mu,
                                                  const float* __restrict__ sg,
                                                  const float* __restrict__ wmu,
                                                  const float* __restrict__ bmu,
                                                  const float* __restrict__ wsg,
                                                  const float* __restrict__ bsg,
                                                  float* __restrict__ out) {
    int r = blockIdx.x * blockDim.x + threadIdx.x;
    float smu = bmu[0], ssg = bsg[0];
    #pragma unroll
    for (int t = 0; t < TT; ++t) {
        smu += mu[(size_t)r * TT + t] * wmu[t];
        ssg += sg[(size_t)r * TT + t] * wsg[t];
    }
    smu *= (1.0f / 3.0f);
    ssg *= (1.0f / 3.0f);
    #pragma unroll
    for (int d = 0; d < DD; ++d) {
        out[(size_t)r * DD + d]       = smu;
        out[(size_t)LNM + r * DD + d] = ssg;
    }
}

// ---------------------------------------------------------------------------
// host-side GAT driver
// ---------------------------------------------------------------------------
struct GatScratch {
    float*    ft;       // NB*48
    float*    outacc;   // NB*40 (contiguous with emax/esum for one big memset)
    unsigned* emaxk;    // NB*4
    float*    esum;     // NB*4
    float*    Bmat;     // 12*48
    float*    red;      // 2
    float*    tmp;      // LNM
};

static void run_gat(hipStream_t stream, const GatScratch& S,
                    const float* feat, const float* fc, const float* al, const float* ar,
                    const int* src, const int* dst,
                    float* lnDst /* nullptr for add mode */, float* addDst) {
    k_prep_w<<<1, 64, 0, stream>>>(fc, al, ar, S.Bmat, S.red);
    // outacc (12.8M) + emax (1.28M) + esum (1.28M) contiguous: one vector memset
    constexpr int Z4 = (NB * 40 + 2 * NB * HH) / 4;     // 3,840,000 float4
    k_zero4<<<Z4 / 256, 256, 0, stream>>>((float4*)S.outacc, Z4);
    k_gemm<<<NB / 64, 128, 0, stream>>>(feat, S.Bmat, S.ft);
    k_edge_max<<<EE * BB / 256, 256, 0, stream>>>(S.ft, src, dst, S.emaxk);
    k_edge_sum<<<EE * BB / 256, 256, 0, stream>>>(S.ft, src, dst, S.emaxk, S.esum);
    k_edge_scatter<<<EE * BB / 256, 256, 0, stream>>>(S.ft, src, dst, S.emaxk, S.esum, S.outacc);
    if (lnDst) {
        k_hm_reduce<<<NB / 256, 256, 0, stream>>>(S.outacc, S.tmp, S.red);
        k_ln_apply<<<LNM / 4 / 256, 256, 0, stream>>>((const float4*)S.tmp, S.red, (float4*)lnDst);
    } else {
        k_hm_addout<<<NB / 256, 256, 0, stream>>>(S.outacc, addDst);
    }
}

extern "C" void kernel_launch(void* const* d_in, const int* in_sizes, int n_in,
                              void* d_out, int out_size, void* d_ws, size_t ws_size,
                              hipStream_t stream) {
    // inputs in setup_inputs() dict order
    const float* x       = (const float*)d_in[0];
    const int*   ps_src  = (const int*)d_in[1];
    const int*   ps_dst  = (const int*)d_in[2];
    const int*   rl_src  = (const int*)d_in[3];
    const int*   rl_dst  = (const int*)d_in[4];
    const float* mu_p_fc = (const float*)d_in[5];
    const float* mu_p_al = (const float*)d_in[6];
    const float* mu_p_ar = (const float*)d_in[7];
    const float* sg_p_fc = (const float*)d_in[8];
    const float* sg_p_al = (const float*)d_in[9];
    const float* sg_p_ar = (const float*)d_in[10];
    const float* mu_r_fc = (const float*)d_in[11];
    const float* mu_r_al = (const float*)d_in[12];
    const float* mu_r_ar = (const float*)d_in[13];
    const float* sg_r_fc = (const float*)d_in[14];
    const float* sg_r_al = (const float*)d_in[15];
    const float* sg_r_ar = (const float*)d_in[16];
    const float* reg_mu_w = (const float*)d_in[17];
    const float* reg_mu_b = (const float*)d_in[18];
    const float* reg_sg_w = (const float*)d_in[19];
    const float* reg_sg_b = (const float*)d_in[20];
    float* out = (float*)d_out;

    // workspace layout (floats); all offsets are multiples of 16 bytes
    float* wsf = (float*)d_ws;
    size_t off = 0;
    float* mu     = wsf + off; off += (size_t)NB * TT;        // 3.2M
    float* sigma  = wsf + off; off += (size_t)NB * TT;        // 3.2M
    float* ft     = wsf + off; off += (size_t)NB * FTC;       // 15.36M
    float* outacc = wsf + off; off += (size_t)NB * HH * DD;   // 12.8M
    unsigned* emaxk = (unsigned*)(wsf + off); off += (size_t)NB * HH; // 1.28M (contig after outacc)
    float* esum   = wsf + off; off += (size_t)NB * HH;        // 1.28M (contig after emax)
    float* tmp    = wsf + off; off += (size_t)LNM;            // 3.2M
    float* bufA   = wsf + off; off += (size_t)LNM;
    float* bufB   = wsf + off; off += (size_t)LNM;
    float* h_p    = wsf + off; off += (size_t)LNM;
    float* h_rl   = wsf + off; off += (size_t)LNM;
    float* Bmat   = wsf + off; off += 12 * FTC;
    float* red    = wsf + off; off += 2;

    GatScratch S{ft, outacc, emaxk, esum, Bmat, red, tmp};

    // layer offsets within (L, 40, 10) / (L, 4, 10) params
    const int FCL = HH * DD * TT;  // 400
    const int AL  = HH * DD;       // 40

    // unpack x -> mu, sigma
    k_unpack<<<NB * TT / 256, 256, 0, stream>>>(x, mu, sigma);

    // pseudo-graph alignment layer 0
    run_gat(stream, S, mu,    mu_p_fc,       mu_p_al,      mu_p_ar,      ps_src, ps_dst, bufA, nullptr);
    run_gat(stream, S, sigma, sg_p_fc,       sg_p_al,      sg_p_ar,      ps_src, ps_dst, bufB, nullptr);
    k_combine<<<LNM / 4 / 256, 256, 0, stream>>>((const float4*)bufA, (const float4*)bufB, (float4*)h_p);

    // real-graph alignment layer 0
    run_gat(stream, S, mu,    mu_r_fc,       mu_r_al,      mu_r_ar,      rl_src, rl_dst, bufA, nullptr);
    run_gat(stream, S, sigma, sg_r_fc,       sg_r_al,      sg_r_ar,      rl_src, rl_dst, bufB, nullptr);
    k_combine<<<LNM / 4 / 256, 256, 0, stream>>>((const float4*)bufA, (const float4*)bufB, (float4*)h_rl);

    // regression heads /3 initialize output (overwrites poison)
    k_init_out<<<NB / 256, 256, 0, stream>>>(mu, sigma, reg_mu_w, reg_mu_b, reg_sg_w, reg_sg_b, out);

    // final GAT layers, accumulated /3 into output halves
    run_gat(stream, S, h_p,  mu_p_fc + FCL, mu_p_al + AL, mu_p_ar + AL, ps_src, ps_dst, nullptr, out);
    run_gat(stream, S, h_rl, mu_r_fc + FCL, mu_r_al + AL, mu_r_ar + AL, rl_src, rl_dst, nullptr, out);
    run_gat(stream, S, h_p,  sg_p_fc + FCL, sg_p_al + AL, sg_p_ar + AL, ps_src, ps_dst, nullptr, out + LNM);
    run_gat(stream, S, h_rl, sg_r_fc + FCL, sg_r_al + AL, sg_r_ar + AL, rl_src, rl_dst, nullptr, out + LNM);
}